// FaceDetector_78993038508455
// MI455X (gfx1250) — compile-verified
//
#include <hip/hip_runtime.h>
#include <hip/hip_bf16.h>
#include <math.h>

#define CONF_THR 0.02f
#define NMS_THR  0.4f
#define PRE_K    5000
#define POST_K   750
#define KPAD     5024              // PRE_K padded to multiple of 32
#define NW       157               // KPAD / 32 mask words per row
#define CANDCAP  8192              // bitonic sort size (power of 2 >= PRE_K)

struct SelState {
    unsigned long long prefix;     // accumulating 64-bit threshold key
    unsigned remaining;            // how many still to pick inside prefix
    unsigned counter;              // compaction counter
    unsigned keptCount;            // NMS kept count
    unsigned pad;
};

// ---------------------------------------------------------------- init
__global__ void k_init(unsigned* hist, SelState* st) {
    int t = blockIdx.x * blockDim.x + threadIdx.x;
    if (t < 65536) hist[t] = 0u;
    if (t == 0) {
        st->prefix = 0ull;
        st->remaining = PRE_K;
        st->counter = 0u;
        st->keptCount = 0u;
    }
}

// ---------------------------------------------------------------- decode
__global__ void k_decode(const float* __restrict__ loc,
                         const float* __restrict__ conf,
                         const float* __restrict__ lmk,
                         const float* __restrict__ priors,
                         const int* __restrict__ pW, const int* __restrict__ pH,
                         float* __restrict__ boxesF, float* __restrict__ lmkF,
                         float* __restrict__ scoresF,
                         unsigned long long* __restrict__ keys, int n) {
    int i = blockIdx.x * blockDim.x + threadIdx.x;
    if (i >= n) return;
    float W = (float)pW[0], H = (float)pH[0];

    float px = priors[4*i+0], py = priors[4*i+1];
    float pw = priors[4*i+2], ph = priors[4*i+3];
    float lx = loc[4*i+0], ly = loc[4*i+1], lw = loc[4*i+2], lh = loc[4*i+3];

    float cx = px + lx * 0.1f * pw;
    float cy = py + ly * 0.1f * ph;
    float bw = pw * expf(lw * 0.2f);
    float bh = ph * expf(lh * 0.2f);

    boxesF[4*i+0] = (cx - bw * 0.5f) * W;
    boxesF[4*i+1] = (cy - bh * 0.5f) * H;
    boxesF[4*i+2] = (cx + bw * 0.5f) * W;
    boxesF[4*i+3] = (cy + bh * 0.5f) * H;

    #pragma unroll
    for (int t = 0; t < 5; ++t) {
        float ox = px + lmk[10*i + 2*t + 0] * 0.1f * pw;
        float oy = py + lmk[10*i + 2*t + 1] * 0.1f * ph;
        lmkF[10*i + 2*t + 0] = ox * W;
        lmkF[10*i + 2*t + 1] = oy * H;
    }

    float s = conf[2*i + 1];
    scoresF[i] = s;
    unsigned kb = 0u;
    if (s > CONF_THR) kb = __float_as_uint(s) | 0x80000000u; // s>0 => monotone map
    keys[i] = ((unsigned long long)kb << 32) | (unsigned)(~(unsigned)i);
}

// ---------------------------------------------------------------- radix select (4x16-bit)
__global__ void k_hist(const unsigned long long* __restrict__ keys,
                       unsigned* __restrict__ hist,
                       const SelState* __restrict__ st, int n, int round) {
    int i = blockIdx.x * blockDim.x + threadIdx.x;
    if (i >= n) return;
    unsigned long long key = keys[i];
    int shift = 48 - 16 * round;
    bool match = true;
    if (round > 0) {
        int hs = shift + 16;
        match = (key >> hs) == (st->prefix >> hs);
    }
    if (match) atomicAdd(&hist[(unsigned)((key >> shift) & 0xFFFFull)], 1u);
}

__global__ __launch_bounds__(1024) void k_radix_scan(unsigned* __restrict__ hist,
                                                     SelState* __restrict__ st,
                                                     int round) {
    __shared__ unsigned sP[1024];
    int t = threadIdx.x;
    unsigned rem = st->remaining;

    int hi = 65535 - 64 * t;              // descending 64-bucket chunk per thread
    unsigned sum = 0;
    for (int b = hi; b > hi - 64; --b) sum += hist[b];
    sP[t] = sum;
    __syncthreads();
    for (int off = 1; off < 1024; off <<= 1) {   // inclusive scan
        unsigned v = (t >= off) ? sP[t - off] : 0u;
        __syncthreads();
        sP[t] += v;
        __syncthreads();
    }
    unsigned incl = sP[t];
    unsigned excl = incl - sum;
    if (excl < rem && incl >= rem) {             // exactly one winner
        unsigned cum = excl;
        int bucket = hi;
        for (int b = hi; b > hi - 64; --b) {
            cum += hist[b];
            if (cum >= rem) { bucket = b; break; }
        }
        unsigned newrem = rem - (cum - hist[bucket]);
        int shift = 48 - 16 * round;
        st->prefix |= ((unsigned long long)(unsigned)bucket) << shift;
        st->remaining = newrem;
    }
    __syncthreads();
    for (int b = t; b < 65536; b += 1024) hist[b] = 0u;  // ready for next round
}

// ---------------------------------------------------------------- compact candidates
__global__ void k_compact(const unsigned long long* __restrict__ keys,
                          unsigned long long* __restrict__ cand,
                          SelState* __restrict__ st, int n) {
    int i = blockIdx.x * blockDim.x + threadIdx.x;
    if (i >= n) return;
    unsigned long long key = keys[i];
    if (key >= st->prefix) {                     // exactly PRE_K keys (unique keys)
        unsigned pos = atomicAdd(&st->counter, 1u);
        if (pos < CANDCAP) cand[pos] = key;
    }
}

// ---------------------------------------------------------------- bitonic sort (descending)
__global__ __launch_bounds__(1024) void k_sort(const unsigned long long* __restrict__ cand,
                                               const SelState* __restrict__ st,
                                               unsigned* __restrict__ order) {
    __shared__ unsigned long long sK[CANDCAP];   // 64 KB LDS
    int t = threadIdx.x;
    unsigned cnt = st->counter;
    if (cnt > CANDCAP) cnt = CANDCAP;
    for (int k = t; k < CANDCAP; k += 1024) sK[k] = (k < (int)cnt) ? cand[k] : 0ull;
    __syncthreads();

    for (unsigned k = 2; k <= CANDCAP; k <<= 1) {
        for (unsigned j = k >> 1; j > 0; j >>= 1) {
            for (unsigned i = t; i < CANDCAP; i += 1024) {
                unsigned p = i ^ j;
                if (p > i) {
                    unsigned long long a = sK[i], b = sK[p];
                    bool up = ((i & k) == 0);
                    if (up ? (a < b) : (a > b)) { sK[i] = b; sK[p] = a; }
                }
            }
            __syncthreads();
        }
    }
    for (int k = t; k < KPAD; k += 1024) {
        if (k < (int)cnt && k < PRE_K)
            order[k] = ~(unsigned)(sK[k] & 0xFFFFFFFFull);   // recover original index
        else
            order[k] = 0xFFFFFFFFu;                          // sentinel: invalid slot
    }
}

// ---------------------------------------------------------------- gather sorted boxes/scores
__global__ void k_gather(const float* __restrict__ boxesF,
                         const float* __restrict__ scoresF,
                         unsigned* __restrict__ order,
                         float* __restrict__ boxesS,
                         float* __restrict__ scoresS) {
    int k = blockIdx.x * blockDim.x + threadIdx.x;
    if (k >= KPAD) return;
    unsigned oi = order[k];
    bool ok = (k < PRE_K) && (oi != 0xFFFFFFFFu);
    if (ok) {
        scoresS[k] = (scoresF[oi] > CONF_THR) ? scoresF[oi] : -INFINITY;
        boxesS[4*k+0] = boxesF[4*oi+0];
        boxesS[4*k+1] = boxesF[4*oi+1];
        boxesS[4*k+2] = boxesF[4*oi+2];
        boxesS[4*k+3] = boxesF[4*oi+3];
    } else {
        scoresS[k] = -INFINITY;
        boxesS[4*k+0] = 0.f; boxesS[4*k+1] = 0.f;
        boxesS[4*k+2] = 0.f; boxesS[4*k+3] = 0.f;
        order[k] = 0u;                                       // safe index for later reads
    }
}

// ---------------------------------------------------------------- IoU suppression bitmask
// 8 waves per block; each wave owns one upper-triangular 32x32 tile and stages
// its 32 column boxes into its own LDS slice via gfx1250
// GLOBAL_LOAD_ASYNC_TO_LDS_B128 (ASYNCcnt path). No intra-block sync needed.
__global__ __launch_bounds__(256) void k_iou_mask(const float* __restrict__ boxesS,
                                                  unsigned* __restrict__ mask) {
    int lane = threadIdx.x;                    // 0..31
    int wv   = threadIdx.y;                    // 0..7
    int c = blockIdx.x * 8 + wv;               // column tile
    int r = blockIdx.y;                        // row tile
    if (c >= NW || c < r) return;              // only j >= i tiles ever read

    __shared__ float sCol[8][32 * 4];

    // Async copy of this tile's 32 column boxes (16B per lane) into LDS.
    unsigned ldsOff = (unsigned)(uintptr_t)(&sCol[wv][lane * 4]);
    unsigned long long ga =
        (unsigned long long)(uintptr_t)(boxesS + (c * 32 + lane) * 4);
    asm volatile("global_load_async_to_lds_b128 %0, %1, off"
                 :: "v"(ldsOff), "v"(ga) : "memory");

    int i = r * 32 + lane;
    float4 rb = reinterpret_cast<const float4*>(boxesS)[i];
    float ax1 = rb.x, ay1 = rb.y, ax2 = rb.z, ay2 = rb.w;
    float areaA = (ax2 - ax1) * (ay2 - ay1);

    asm volatile("s_wait_asynccnt 0" ::: "memory");

    unsigned bits = 0u;
    #pragma unroll 4
    for (int jj = 0; jj < 32; ++jj) {
        int j = c * 32 + jj;
        float bx1 = sCol[wv][jj*4+0], by1 = sCol[wv][jj*4+1];
        float bx2 = sCol[wv][jj*4+2], by2 = sCol[wv][jj*4+3];
        float ix1 = fmaxf(ax1, bx1), iy1 = fmaxf(ay1, by1);
        float ix2 = fminf(ax2, bx2), iy2 = fminf(ay2, by2);
        float iw = fmaxf(ix2 - ix1, 0.f), ih = fmaxf(iy2 - iy1, 0.f);
        float inter = iw * ih;
        float areaB = (bx2 - bx1) * (by2 - by1);
        float uni = fmaxf(areaA + areaB - inter, 1e-12f);
        float iou = inter / uni;
        if ((iou > NMS_THR) && (j > i)) bits |= (1u << jj);
    }
    mask[i * NW + c] = bits;
}

// ---------------------------------------------------------------- greedy NMS gather
// Word-block formulation: 157 outer iterations instead of 5000. Per 32-row
// block: prefetch diagonal mask words + scores to LDS in parallel, one lane
// resolves keep bits serially from LDS, then all threads OR the kept rows'
// masks into removed[] with coalesced reads.
__global__ __launch_bounds__(160) void k_nms_seq(const unsigned* __restrict__ mask,
                                                 const float* __restrict__ scoresS,
                                                 unsigned* __restrict__ kept,
                                                 SelState* __restrict__ st) {
    __shared__ unsigned removed[NW];
    __shared__ unsigned sDiag[32];
    __shared__ float    sSc[32];
    __shared__ unsigned sKeepBits;
    __shared__ int      sCount;
    int t = threadIdx.x;
    if (t < NW) removed[t] = 0u;
    if (t == 0) sCount = 0;
    __syncthreads();

    for (int wb = 0; wb < NW; ++wb) {
        int i0 = wb * 32;
        if (t < 32) {
            sDiag[t] = mask[(i0 + t) * NW + wb];   // intra-word bits, only j > i
            sSc[t]   = scoresS[i0 + t];
        }
        __syncthreads();
        if (t == 0) {
            unsigned rem = removed[wb];
            unsigned kb = 0u;
            int lim = (PRE_K - i0 < 32) ? (PRE_K - i0) : 32;
            int cnt = sCount;
            for (int b = 0; b < lim; ++b) {
                if ((sSc[b] > CONF_THR) && !((rem >> b) & 1u)) {
                    kb |= (1u << b);
                    rem |= sDiag[b];
                    if (cnt < POST_K) kept[cnt] = (unsigned)(i0 + b);
                    cnt++;
                }
            }
            sCount = cnt;
            sKeepBits = kb;
        }
        __syncthreads();
        unsigned kb = sKeepBits;
        if (kb) {
            for (int w = wb + 1 + t; w < NW; w += (int)blockDim.x) {
                unsigned acc = removed[w];
                unsigned m = kb;
                while (m) {
                    int b = __ffs(m) - 1;
                    m &= m - 1u;
                    acc |= mask[(i0 + b) * NW + w];
                }
                removed[w] = acc;
            }
        }
        __syncthreads();
    }
    if (t == 0) st->keptCount = (unsigned)((sCount < POST_K) ? sCount : POST_K);
}

// ---------------------------------------------------------------- output scatter
__global__ void k_output(const float* __restrict__ boxesS,
                         const float* __restrict__ scoresS,
                         const float* __restrict__ lmkF,
                         const unsigned* __restrict__ order,
                         const unsigned* __restrict__ kept,
                         const SelState* __restrict__ st,
                         float* __restrict__ out) {
    int idx = blockIdx.x * blockDim.x + threadIdx.x;
    if (idx >= POST_K * 15) return;
    int r = idx / 15, c = idx % 15;
    float v = 0.f;
    unsigned cnt = st->keptCount;
    if ((unsigned)r < cnt) {
        unsigned k = kept[r];
        if (c < 4)       v = boxesS[k * 4 + c];
        else if (c == 4) v = scoresS[k];
        else             v = lmkF[order[k] * 10 + (c - 5)];
    }
    out[idx] = v;
}

// ---------------------------------------------------------------- launch
extern "C" void kernel_launch(void* const* d_in, const int* in_sizes, int n_in,
                              void* d_out, int out_size, void* d_ws, size_t ws_size,
                              hipStream_t stream) {
    const float* loc    = (const float*)d_in[0];
    const float* conf   = (const float*)d_in[1];
    const float* lmk    = (const float*)d_in[2];
    const float* priors = (const float*)d_in[3];
    const int*   pW     = (const int*)d_in[4];
    const int*   pH     = (const int*)d_in[5];
    float* out = (float*)d_out;

    const int n = in_sizes[0] / 4;

    auto al = [](size_t x) { return (x + 255) & ~(size_t)255; };
    char* ws = (char*)d_ws;
    size_t o = 0;
    size_t oKeys  = o; o = al(o + (size_t)n * 8);
    size_t oBoxF  = o; o = al(o + (size_t)n * 4 * 4);
    size_t oLmkF  = o; o = al(o + (size_t)n * 10 * 4);
    size_t oScF   = o; o = al(o + (size_t)n * 4);
    size_t oHist  = o; o = al(o + 65536 * 4);
    size_t oState = o; o = al(o + 64);
    size_t oCand  = o; o = al(o + (size_t)CANDCAP * 8);
    size_t oOrder = o; o = al(o + (size_t)KPAD * 4);
    size_t oBoxS  = o; o = al(o + (size_t)KPAD * 4 * 4);
    size_t oScS   = o; o = al(o + (size_t)KPAD * 4);
    size_t oMask  = o; o = al(o + (size_t)KPAD * NW * 4);
    size_t oKept  = o; o = al(o + (size_t)POST_K * 4);
    (void)ws_size;

    unsigned long long* keys = (unsigned long long*)(ws + oKeys);
    float*    boxesF = (float*)(ws + oBoxF);
    float*    lmkF   = (float*)(ws + oLmkF);
    float*    scF    = (float*)(ws + oScF);
    unsigned* hist   = (unsigned*)(ws + oHist);
    SelState* st     = (SelState*)(ws + oState);
    unsigned long long* cand = (unsigned long long*)(ws + oCand);
    unsigned* order  = (unsigned*)(ws + oOrder);
    float*    boxesS = (float*)(ws + oBoxS);
    float*    scS    = (float*)(ws + oScS);
    unsigned* mask   = (unsigned*)(ws + oMask);
    unsigned* kept   = (unsigned*)(ws + oKept);

    int gN = (n + 255) / 256;

    k_init<<<(65536 + 255) / 256, 256, 0, stream>>>(hist, st);
    k_decode<<<gN, 256, 0, stream>>>(loc, conf, lmk, priors, pW, pH,
                                     boxesF, lmkF, scF, keys, n);
    for (int r = 0; r < 4; ++r) {
        k_hist<<<gN, 256, 0, stream>>>(keys, hist, st, n, r);
        k_radix_scan<<<1, 1024, 0, stream>>>(hist, st, r);
    }
    k_compact<<<gN, 256, 0, stream>>>(keys, cand, st, n);
    k_sort<<<1, 1024, 0, stream>>>(cand, st, order);
    k_gather<<<(KPAD + 255) / 256, 256, 0, stream>>>(boxesF, scF, order, boxesS, scS);
    k_iou_mask<<<dim3((NW + 7) / 8, NW), dim3(32, 8), 0, stream>>>(boxesS, mask);
    k_nms_seq<<<1, 160, 0, stream>>>(mask, scS, kept, st);
    k_output<<<(POST_K * 15 + 255) / 256, 256, 0, stream>>>(
        boxesS, scS, lmkF, order, kept, st, out);
    (void)out_size; (void)n_in;
}